// SSIM3d_5643587027073
// MI455X (gfx1250) — compile-verified
//
#include <hip/hip_runtime.h>

// ---------------------------------------------------------------------------
// Fused 3D-SSIM for MI455X (gfx1250), round 3.
//  - one kernel, tile-resident in LDS (no global intermediates)
//  - separable 11-tap Gaussian as chains of V_WMMA_F32_16X16X4_F32
//  - intermediate tiles stored in WMMA C-layout-native order: stores are
//    base + immediate offsets (no clamps), reads use a cheap hoistable map
//  - wave id forced scalar (readfirstlane) -> scalar task loops, EXEC
//    stays all-ones around every WMMA
// Output tile per block: 8(z) x 16(y) x 16(x); halo +10 each dim.
// ---------------------------------------------------------------------------

typedef float v2f __attribute__((ext_vector_type(2)));
typedef float v8f __attribute__((ext_vector_type(8)));

#define TZ 8
#define TY 16
#define TX 16
#define HZ (TZ + 10)   // 18
#define HY (TY + 10)   // 26
#define HX (TX + 10)   // 26
#define DIMX 192

#define RAW_Y 27                           // rows 0..25 data, row 26 zero
#define RAW_X 28                           // cols 0..25 data, 26..27 zero
#define RAW_ELEMS (HZ * RAW_Y * RAW_X)     // 13608
#define TWP_ELEMS (HZ * 2 * 8 * 32)        // 9216  (z, ygroup, vgpr, lane)
#define THP_Z 20                           // rows 0..17 data, 18..19 zero
#define THP_ELEMS (THP_Z * 8 * 32)         // 5120  (z, vgpr, lane)
#define SMEM_FLOATS (2 * RAW_ELEMS + TWP_ELEMS + THP_ELEMS + 8 + 12)

// Normalized Gaussian, win=11, sigma=1.5 (matches reference _gauss1d()).
__constant__ float KW[11] = {
    1.0283804e-03f, 7.5987587e-03f, 3.6000773e-02f, 1.0936069e-01f,
    2.1300553e-01f, 2.6601172e-01f, 2.1300553e-01f, 1.0936069e-01f,
    3.6000773e-02f, 7.5987587e-03f, 1.0283804e-03f};

__device__ __forceinline__ v8f wmma_f32(v2f a, v2f b, v8f c) {
  return __builtin_amdgcn_wmma_f32_16x16x4_f32(false, a, false, b, (short)0, c,
                                               false, false);
}

__device__ __forceinline__ v8f vzero8() {
  v8f z = {0.f, 0.f, 0.f, 0.f, 0.f, 0.f, 0.f, 0.f};
  return z;
}

// C-layout slot of matrix element (row m, col n) within one 16x16 tile:
// lanes 0-15 hold m=0..7 (vgpr v=m), lanes 16-31 hold m=8..15.
__device__ __forceinline__ int cslot(int m) {
  return ((m & 7) << 5) + ((m & 8) << 1);  // v*32 + hi*16  (col added by caller)
}

// Load one A-fragment pair (x-contiguous) for field F from the raw tiles.
template <int F>
__device__ __forceinline__ v2f ldA(const float* rowA, const float* rowB,
                                   int j0) {
  if (F == 0) return *(const v2f*)(rowA + j0);
  if (F == 1) return *(const v2f*)(rowB + j0);
  if (F == 2) { v2f a = *(const v2f*)(rowA + j0); return a * a; }
  if (F == 3) { v2f b = *(const v2f*)(rowB + j0); return b * b; }
  v2f a = *(const v2f*)(rowA + j0);
  v2f b = *(const v2f*)(rowB + j0);
  return a * b;
}

// Three blur passes for one field F. Result: r0 (y = wave), r1 (y = wave+8).
// C-layout: lanes 0-15 hold z = vgpr idx (valid, z<8); lanes 16-31 discarded.
template <int F>
__device__ __forceinline__ void field_pass(const float* rawA, const float* rawB,
                                           float* twp, float* thp,
                                           const v2f* wf, int wave, int lane,
                                           v8f& r0, v8f& r1) {
  const int n16 = lane & 15;
  const int khalf = (lane < 16) ? 0 : 2;   // A/B K split across half-waves

  // ---- pass X: OUT(16y x 16x) = IN(16y x 28j) * Wx(28j x 16x) ----
  for (int t = wave; t < HZ * 2; t += 8) {
    int z = t >> 1, yg = t & 1;
    int m = yg * 16 + n16;
    m = (m < RAW_Y - 1) ? m : (RAW_Y - 1);        // clamp to zero row 26
    const float* rowA = rawA + (z * RAW_Y + m) * RAW_X;
    const float* rowB = rawB + (z * RAW_Y + m) * RAW_X;
    v8f acc = vzero8();
#pragma unroll
    for (int c = 0; c < 7; ++c) {
      acc = wmma_f32(ldA<F>(rowA, rowB, 4 * c + khalf), wf[c], acc);
    }
    float* dst = twp + (z * 2 + yg) * 256 + lane;  // native C-layout store
#pragma unroll
    for (int v = 0; v < 8; ++v) dst[v * 32] = acc[v];
  }
  __syncthreads();

  // ---- pass Y: OUT(16yo x 16x) = Wy(16yo x 28j) * IN(28j x 16x) ----
  for (int t = wave; t < HZ; t += 8) {
    const float* src = twp + t * 512 + n16;
    v8f acc = vzero8();
#pragma unroll
    for (int c = 0; c < 7; ++c) {
      int j0 = 4 * c + khalf;
      int j1 = j0 + 1;
      v2f b;
      b.x = src[((j0 >> 4) << 8) + cslot(j0 & 15)];
      b.y = src[((j1 >> 4) << 8) + cslot(j1 & 15)];
      acc = wmma_f32(wf[c], b, acc);
    }
    float* dst = thp + t * 256 + lane;             // native C-layout store
#pragma unroll
    for (int v = 0; v < 8; ++v) dst[v * 32] = acc[v];
  }
  __syncthreads();

  // ---- pass Z: OUT(16zo x 16x) = Wz(16zo x 20j) * IN(20j x 16x) ----
#pragma unroll
  for (int ti = 0; ti < 2; ++ti) {
    // value (z=j, y=wave+8*ti, x=n16) lives at j*256 + wave*32 + ti*16 + n16
    const float* src = thp + khalf * 256 + wave * 32 + ti * 16 + n16;
    v8f acc = vzero8();
#pragma unroll
    for (int c = 0; c < 5; ++c) {
      v2f b;
      b.x = src[c * 1024];
      b.y = src[c * 1024 + 256];
      acc = wmma_f32(wf[c], b, acc);
    }
    if (ti == 0) r0 = acc; else r1 = acc;
  }
  __syncthreads();
}

__device__ __forceinline__ float ssim_px(float mu1, float mu2, float ex2,
                                         float ey2, float exy) {
  const float C1 = 1.0e-4f, C2 = 9.0e-4f;
  float mu1s = mu1 * mu1, mu2s = mu2 * mu2, m12 = mu1 * mu2;
  float s1 = ex2 - mu1s;
  float s2 = ey2 - mu2s;
  float s12 = exy - m12;
  float num = (2.0f * m12 + C1) * (2.0f * s12 + C2);
  float den = (mu1s + mu2s + C1) * (s1 + s2 + C2);
  return num / den;
}

__global__ __launch_bounds__(256, 1) void ssim3d_wmma_kernel(
    const float* __restrict__ img1, const float* __restrict__ img2,
    float* __restrict__ out) {
  extern __shared__ float smem[];
  float* rawA = smem;                        // [18][27][28]
  float* rawB = rawA + RAW_ELEMS;            // [18][27][28]
  float* twp = rawB + RAW_ELEMS;             // [18][2][8][32] C-layout tiles
  float* thp = twp + TWP_ELEMS;              // [20][8][32]    C-layout tiles
  float* red = thp + THP_ELEMS;              // [8]
  float* kwt = red + 8;                      // [12], kwt[11] = 0

  const int tid = threadIdx.x;
  const int lane = tid & 31;
  const int wave = __builtin_amdgcn_readfirstlane(tid >> 5);  // scalar

  // Decode tile coords.
  const int tilesX = DIMX / TX, tilesY = DIMX / TY, tilesZ = DIMX / TZ;
  const int perBatch = tilesX * tilesY * tilesZ;
  int bid = blockIdx.x;
  int batch = bid / perBatch;
  int r = bid - batch * perBatch;
  int tz = r / (tilesX * tilesY);
  int r2 = r - tz * (tilesX * tilesY);
  int ty = r2 / tilesX;
  int tx = r2 - ty * tilesX;
  const int z0 = tz * TZ, y0 = ty * TY, x0 = tx * TX;

  // ---- init: weight table, zero pad rows of thp (z=18,19) ----
  if (tid < 12) kwt[tid] = (tid < 11) ? KW[tid] : 0.0f;
  thp[HZ * 256 + tid] = 0.0f;
  thp[HZ * 256 + 256 + tid] = 0.0f;

  // ---- stage padded halo tiles of both images into LDS ----
  long base = (long)batch * ((long)DIMX * DIMX * DIMX);
  for (int i = tid; i < RAW_ELEMS; i += 256) {
    int z = i / (RAW_Y * RAW_X);
    int rr = i - z * (RAW_Y * RAW_X);
    int y = rr / RAW_X;
    int x = rr - y * RAW_X;
    int gz = z0 - 5 + z, gy = y0 - 5 + y, gx = x0 - 5 + x;
    float va = 0.0f, vb = 0.0f;
    if (y < HY && x < HX && (unsigned)gz < (unsigned)DIMX &&
        (unsigned)gy < (unsigned)DIMX && (unsigned)gx < (unsigned)DIMX) {
      long idx = base + (long)gz * (DIMX * DIMX) + gy * DIMX + gx;
      va = img1[idx];
      vb = img2[idx];
    }
    rawA[i] = va;
    rawB[i] = vb;
  }
  __syncthreads();

  // ---- per-lane Gaussian fragments (shared by all passes/fields) ----
  const int n16 = lane & 15;
  const int khalf = (lane < 16) ? 0 : 2;
  v2f wf[7];
#pragma unroll
  for (int c = 0; c < 7; ++c) {
    int j0 = 4 * c + khalf;
    unsigned d0 = (unsigned)(j0 - n16);
    unsigned d1 = (unsigned)(j0 + 1 - n16);
    wf[c].x = kwt[d0 < 11u ? d0 : 11u];
    wf[c].y = kwt[d1 < 11u ? d1 : 11u];
  }

  // ---- blur the five fields through the matrix pipe ----
  v8f mu1a, mu1b, mu2a, mu2b, x2a, x2b, y2a, y2b, xya, xyb;
  field_pass<0>(rawA, rawB, twp, thp, wf, wave, lane, mu1a, mu1b);
  field_pass<1>(rawA, rawB, twp, thp, wf, wave, lane, mu2a, mu2b);
  field_pass<2>(rawA, rawB, twp, thp, wf, wave, lane, x2a, x2b);
  field_pass<3>(rawA, rawB, twp, thp, wf, wave, lane, y2a, y2b);
  field_pass<4>(rawA, rawB, twp, thp, wf, wave, lane, xya, xyb);

  // ---- SSIM map + mean ----
  float sum = 0.0f;
  if (lane < 16) {  // lanes 16-31 hold z rows 8..15, outside the 8-deep tile
#pragma unroll
    for (int v = 0; v < 8; ++v) {
      sum += ssim_px(mu1a[v], mu2a[v], x2a[v], y2a[v], xya[v]);
      sum += ssim_px(mu1b[v], mu2b[v], x2b[v], y2b[v], xyb[v]);
    }
  }
#pragma unroll
  for (int off = 16; off > 0; off >>= 1) sum += __shfl_down(sum, off, 32);
  if (lane == 0) red[wave] = sum;
  __syncthreads();
  if (tid == 0) {
    float tot = 0.0f;
#pragma unroll
    for (int w = 0; w < 8; ++w) tot += red[w];
    const float invN = 1.0f / (2.0f * 192.0f * 192.0f * 192.0f);
    atomicAdd(out, tot * invN);
  }
}

extern "C" void kernel_launch(void* const* d_in, const int* in_sizes, int n_in,
                              void* d_out, int out_size, void* d_ws,
                              size_t ws_size, hipStream_t stream) {
  (void)in_sizes; (void)n_in; (void)out_size; (void)d_ws; (void)ws_size;
  const float* img1 = (const float*)d_in[0];
  const float* img2 = (const float*)d_in[1];
  float* out = (float*)d_out;

  const size_t smem_bytes = (size_t)SMEM_FLOATS * sizeof(float);  // ~162 KB
  hipFuncSetAttribute(reinterpret_cast<const void*>(ssim3d_wmma_kernel),
                      hipFuncAttributeMaxDynamicSharedMemorySize,
                      (int)smem_bytes);

  hipMemsetAsync(d_out, 0, sizeof(float), stream);  // mean accumulator

  const int tiles = 2 * (DIMX / TZ) * (DIMX / TY) * (DIMX / TX);  // 6912
  ssim3d_wmma_kernel<<<dim3(tiles), dim3(256), smem_bytes, stream>>>(img1, img2,
                                                                     out);
}